// GCNConv_37349035606524
// MI455X (gfx1250) — compile-verified
//
#include <hip/hip_runtime.h>
#include <hip/hip_bf16.h>

typedef float v2f __attribute__((ext_vector_type(2)));
typedef float v8f __attribute__((ext_vector_type(8)));

#define NNODES 100000
#define FDIM   128

// ---------------------------------------------------------------------------
// GEMM: out[strip*16 .. strip*16+15][0..127] = X_strip @ H   (H is 128x128)
// One wave per 16-row strip. H staged in LDS as paired rows:
//   Hpair[p][n] = (H[2p][n], H[2p+1][n])   -> 64 x 128 float2 = 64KB
// WMMA f32 16x16x4 fragment layouts (ISA 7.12.2):
//   A: lane l holds A[l&15][2*(l>>4)+i] in vgpr i  (two consecutive k -> b64 from X)
//   B: lane l holds B[2*(l>>4)+i][l&15] in vgpr i  (one ds_load_b64 from Hpair)
//   C/D: vgpr i -> row (i + 8*(l>>4)), col (l&15)
// ---------------------------------------------------------------------------
__global__ __launch_bounds__(256) void gemm_xh(const float* __restrict__ X,
                                               const float* __restrict__ H,
                                               float* __restrict__ out,
                                               int nStrips) {
  __shared__ float hl[64 * 128 * 2];  // 64KB of the 320KB WGP LDS

  // cooperative load of H into paired-row LDS layout
  for (int i = threadIdx.x; i < 128 * 128; i += 256) {
    int r = i >> 7;
    int n = i & 127;
    hl[(((r >> 1) << 7) + n) * 2 + (r & 1)] = H[i];
  }
  __syncthreads();

  const int wave  = threadIdx.x >> 5;
  const int lane  = threadIdx.x & 31;
  const int strip = blockIdx.x * 8 + wave;
  if (strip >= nStrips) return;  // wave-uniform exit: EXEC all-1s for WMMA

  const int m    = lane & 15;
  const int half = lane >> 4;  // 0 or 1
  const int row0 = strip * 16;
  const float* xrow = X + (size_t)(row0 + m) * FDIM;

  v8f acc[8];
#pragma unroll
  for (int j = 0; j < 8; ++j) acc[j] = (v8f){0.f, 0.f, 0.f, 0.f, 0.f, 0.f, 0.f, 0.f};

#pragma unroll 4
  for (int kk = 0; kk < 32; ++kk) {
    // A fragment: X[row0+m][kk*4 + 2*half + {0,1}]  (8B-aligned global_load_b64)
    v2f a = *(const v2f*)(xrow + kk * 4 + half * 2);
    const float* hbase = hl + (size_t)(kk * 2 + half) * 128 * 2;
#pragma unroll
    for (int j = 0; j < 8; ++j) {
      // B fragment: (H[k][j*16+m], H[k+1][j*16+m]), k = kk*4 + 2*half, via one ds_load_b64
      v2f b = *(const v2f*)(hbase + (j * 16 + m) * 2);
      acc[j] = __builtin_amdgcn_wmma_f32_16x16x4_f32(
          /*neg_a=*/false, a, /*neg_b=*/false, b,
          /*c_mod=*/(short)0, acc[j], /*reuse_a=*/false, /*reuse_b=*/false);
    }
  }

  // store: acc[j][i] -> out[row0 + i + 8*half][j*16 + m]
#pragma unroll
  for (int j = 0; j < 8; ++j) {
    float* o = out + (size_t)(row0 + 8 * half) * FDIM + j * 16 + m;
#pragma unroll
    for (int i = 0; i < 8; ++i) o[(size_t)i * FDIM] = acc[j][i];
  }
}

// ---------------------------------------------------------------------------
// SpMM scatter: tout[rows[e]] += vals[e] * tin[cols[e]]   (128 floats / edge)
// One wave per edge; 4 floats per lane; L2-resident atomics (51MB << 192MB L2).
// ---------------------------------------------------------------------------
__global__ __launch_bounds__(256) void spmm_atomic(const int* __restrict__ rows,
                                                   const int* __restrict__ cols,
                                                   const float* __restrict__ vals,
                                                   const float* __restrict__ tin,
                                                   float* __restrict__ tout,
                                                   int nEdges) {
  const int edge = blockIdx.x * 8 + (threadIdx.x >> 5);
  if (edge >= nEdges) return;
  const int lane = threadIdx.x & 31;

  const int   r = rows[edge];
  const int   c = cols[edge];
  const float v = vals[edge];

  const float4 d = ((const float4*)(tin + (size_t)c * FDIM))[lane];
  float* dst = tout + (size_t)r * FDIM + lane * 4;
  atomicAdd(dst + 0, v * d.x);
  atomicAdd(dst + 1, v * d.y);
  atomicAdd(dst + 2, v * d.z);
  atomicAdd(dst + 3, v * d.w);
}

// ---------------------------------------------------------------------------
// out = X@H0 + A@(X@H1) + A@(A@(X@H2))
// ---------------------------------------------------------------------------
extern "C" void kernel_launch(void* const* d_in, const int* in_sizes, int n_in,
                              void* d_out, int out_size, void* d_ws, size_t ws_size,
                              hipStream_t stream) {
  const int*   edge_rows = (const int*)d_in[0];
  const int*   edge_cols = (const int*)d_in[1];
  const float* edge_vals = (const float*)d_in[2];
  const float* X         = (const float*)d_in[3];
  const float* H         = (const float*)d_in[4];
  float*       out       = (float*)d_out;

  const int nEdges = in_sizes[0];
  const size_t mat = (size_t)NNODES * FDIM;  // 12.8M floats (51.2 MB)
  float* bufA = (float*)d_ws;
  float* bufB = bufA + mat;

  const int nStrips    = NNODES / 16;              // 6250 (exact)
  const int gemmBlocks = (nStrips + 7) / 8;        // 8 waves/block
  const int spmmBlocks = (nEdges + 7) / 8;

  // bufA = X @ H2
  gemm_xh<<<gemmBlocks, 256, 0, stream>>>(X, H + 2 * 128 * 128, bufA, nStrips);
  // bufB = 0 ; bufB = A @ bufA        (= A @ (X@H2))
  hipMemsetAsync(bufB, 0, mat * sizeof(float), stream);
  spmm_atomic<<<spmmBlocks, 256, 0, stream>>>(edge_rows, edge_cols, edge_vals,
                                              bufA, bufB, nEdges);
  // bufA = X @ H1   (previous spmm read of bufA is ordered before on stream)
  gemm_xh<<<gemmBlocks, 256, 0, stream>>>(X, H + 1 * 128 * 128, bufA, nStrips);
  // out = X @ H0
  gemm_xh<<<gemmBlocks, 256, 0, stream>>>(X, H + 0, out, nStrips);
  // out += A @ bufA   (= A @ (X@H1))
  spmm_atomic<<<spmmBlocks, 256, 0, stream>>>(edge_rows, edge_cols, edge_vals,
                                              bufA, out, nEdges);
  // out += A @ bufB   (= A^2 @ (X@H2))
  spmm_atomic<<<spmmBlocks, 256, 0, stream>>>(edge_rows, edge_cols, edge_vals,
                                              bufB, out, nEdges);
}